// Blobber_88321707475414
// MI455X (gfx1250) — compile-verified
//
#include <hip/hip_runtime.h>

// Fused Blobber for MI455X (gfx1250, wave32).
// out = sig2( box3( sig1( box3(in) ) ) ), only the last reference iteration matters.
// Stage 1: VALU + v_exp_f32 over 66x66 halo region in LDS.
// Stage 2: WMMA f32 16x16x4 band-matrix sandwich P' * W' * Q'.

#define IMG_H 512
#define IMG_W 512
#define BT    64            // output tile per workgroup
#define IN_DIM 68           // BT + 4 (two-conv halo)
#define IN_STRIDE 72
#define A_DIM 66            // BT + 2 (one-conv halo)
#define A_STRIDE 68
#define TMP_STRIDE 36       // 32 + pad

typedef __attribute__((ext_vector_type(2))) float v2f;
typedef __attribute__((ext_vector_type(8))) float v8f;

__device__ __forceinline__ float fast_sigmoid(float x) {
    return 1.0f / (1.0f + __expf(-x));
}

__global__ __launch_bounds__(256) void blobber_fused_kernel(
    const float* __restrict__ in, float* __restrict__ out)
{
    __shared__ float sIn[IN_DIM][IN_STRIDE];
    __shared__ float sA[A_DIM][A_STRIDE];
    __shared__ float sTmp[8][16][TMP_STRIDE];   // per-wave layout-conversion buffer

    const int tid  = threadIdx.x;
    const int lane = tid & 31;
    const int wave = tid >> 5;
    const int m    = lane & 15;   // row index (A-frag) / col index (B-frag, D-frag)
    const int hi   = lane >> 4;   // lane half-group

    const int b    = blockIdx.z;
    const int row0 = blockIdx.y * BT;
    const int col0 = blockIdx.x * BT;
    const float* img = in  + (size_t)b * IMG_H * IMG_W;
    float*       op  = out + (size_t)b * IMG_H * IMG_W;

    // ---- Phase 1: cooperative load 68x68 input patch (zero pad at image border)
    for (int idx = tid; idx < IN_DIM * IN_DIM; idx += 256) {
        int r = idx / IN_DIM, c = idx % IN_DIM;
        int gr = row0 - 2 + r, gc = col0 - 2 + c;
        float v = 0.0f;
        if (gr >= 0 && gr < IMG_H && gc >= 0 && gc < IMG_W)
            v = img[gr * IMG_W + gc];
        sIn[r][c] = v;
    }
    __syncthreads();

    // ---- Phase 2: A = sigmoid((box3(in) - 0.01)*1000) on 66x66 halo region
    const float inv9 = 1.0f / 9.0f;
    for (int idx = tid; idx < A_DIM * A_DIM; idx += 256) {
        int r = idx / A_DIM, c = idx % A_DIM;
        float s = sIn[r  ][c] + sIn[r  ][c+1] + sIn[r  ][c+2]
                + sIn[r+1][c] + sIn[r+1][c+1] + sIn[r+1][c+2]
                + sIn[r+2][c] + sIn[r+2][c+1] + sIn[r+2][c+2];
        sA[r][c] = fast_sigmoid((s * inv9 - 0.01f) * 1000.0f);
    }
    __syncthreads();

    // ---- Phase 3: second box3 via WMMA:  OUT = P'(16x32) * W'(32x32) * Q'(32x16)
    // P'[r][k] = Q'[k][r] = 1/3 for k in {r, r+1, r+2}, else 0 (zero beyond k=17).
    // W'[k][c] = sA[16*ti + k][16*tj + c] (clamped reads hit only zero-weighted terms).
    const float third = 1.0f / 3.0f;
    for (int t = wave; t < 16; t += 8) {
        const int arow = (t >> 2) * 16;   // tile origin in sA coords
        const int acol = (t &  3) * 16;

        // tmp = P' * W'   (two 16x16 D-tiles: window cols 0-15 and 16-31)
        v8f accL = {}; v8f accR = {};
        #pragma unroll
        for (int s = 0; s < 8; ++s) {
            const int k0 = s * 4 + hi * 2;
            v2f a;
            a.x = (k0     >= m && k0     <= m + 2) ? third : 0.0f;
            a.y = (k0 + 1 >= m && k0 + 1 <= m + 2) ? third : 0.0f;
            int r1 = arow + k0;     if (r1 > A_DIM - 1) r1 = A_DIM - 1;
            int r2 = arow + k0 + 1; if (r2 > A_DIM - 1) r2 = A_DIM - 1;
            const int cL = acol + m;
            int cR = acol + m + 16; if (cR > A_DIM - 1) cR = A_DIM - 1;
            v2f bL, bR;
            bL.x = sA[r1][cL]; bL.y = sA[r2][cL];
            bR.x = sA[r1][cR]; bR.y = sA[r2][cR];
            accL = __builtin_amdgcn_wmma_f32_16x16x4_f32(false, a, false, bL,
                                                         (short)0, accL, false, false);
            accR = __builtin_amdgcn_wmma_f32_16x16x4_f32(false, a, false, bR,
                                                         (short)0, accR, false, false);
        }

        // D-layout -> LDS (wave-private), to be re-read in A-layout.
        #pragma unroll
        for (int v = 0; v < 8; ++v) {
            sTmp[wave][v + 8 * hi][m]      = accL[v];
            sTmp[wave][v + 8 * hi][m + 16] = accR[v];
        }
        // Same-wave DS ops are in-order in HW; fence stops compiler reordering.
        asm volatile("" ::: "memory");

        // out_tile = tmp(16x32) * Q'(32x16)
        v8f acc = {};
        #pragma unroll
        for (int s = 0; s < 8; ++s) {
            const int k0 = s * 4 + hi * 2;
            v2f a2, b2;
            a2.x = sTmp[wave][m][k0];
            a2.y = sTmp[wave][m][k0 + 1];
            b2.x = (k0     >= m && k0     <= m + 2) ? third : 0.0f;
            b2.y = (k0 + 1 >= m && k0 + 1 <= m + 2) ? third : 0.0f;
            acc = __builtin_amdgcn_wmma_f32_16x16x4_f32(false, a2, false, b2,
                                                        (short)0, acc, false, false);
        }
        asm volatile("" ::: "memory");

        // sig2 + store (D-layout: row = v + 8*hi, col = m within tile)
        #pragma unroll
        for (int v = 0; v < 8; ++v) {
            const int gr = row0 + arow + v + 8 * hi;
            const int gc = col0 + acol + m;
            op[(size_t)gr * IMG_W + gc] = fast_sigmoid((acc[v] - 0.9f) * 1000.0f);
        }
    }
}

extern "C" void kernel_launch(void* const* d_in, const int* in_sizes, int n_in,
                              void* d_out, int out_size, void* d_ws, size_t ws_size,
                              hipStream_t stream) {
    (void)in_sizes; (void)n_in; (void)d_ws; (void)ws_size; (void)out_size;
    const float* in = (const float*)d_in[0];
    float* out = (float*)d_out;
    dim3 grid(IMG_W / BT, IMG_H / BT, 32);   // 8 x 8 x 32 blocks
    dim3 block(256);                          // 8 wave32 per workgroup
    blobber_fused_kernel<<<grid, block, 0, stream>>>(in, out);
}